// DenseGATLayer_78116865179704
// MI455X (gfx1250) — compile-verified
//
#include <hip/hip_runtime.h>
#include <hip/hip_bf16.h>

// MI455X / gfx1250, wave32.
// Heavy GEMMs: V_WMMA_F32_16X16X4_F32 (exact fp32 matrix path).
// Panel staging: TENSOR_LOAD_TO_LDS (TDM) when available, else cooperative.
// HBM-bound (~400 MB at 23.3 TB/s ~= 18 us); h_e is fully fused (never in HBM).
// Edge GEMM reduced 51.5 -> 17.2 GFLOP by precomputing the broadcast
// new_node_i / new_node_j contributions once per node (eI / eJ).

typedef float v2f __attribute__((ext_vector_type(2)));
typedef float v8f __attribute__((ext_vector_type(8)));

#define B_    2
#define N_    256
#define F_    256      // IN
#define H_    8
#define D_    64
#define HD_   512      // H*D
#define ALPHA_ 0.2f
#define LN_EPS_ 1e-5f

// workspace layout in floats (~9 MiB)
#define WS_XN    0
#define WS_HS    (WS_XN   + B_*N_*F_)
#define WS_HT    (WS_HS   + B_*N_*HD_)
#define WS_SC    (WS_HT   + B_*N_*HD_)
#define WS_NN    (WS_SC   + B_*N_*N_*H_)
#define WS_EI    (WS_NN   + B_*N_*F_)        // nn_i @ Wedge[0:256]   + bedge
#define WS_EJ    (WS_EI   + B_*N_*F_)        // nn_j @ Wedge[256:512]
#define WS_ESUM  (WS_EJ   + B_*N_*F_)
#define WS_ESQ   (WS_ESUM + B_*N_*N_)

// ---------------------------------------------------------------------------
// TDM (Tensor Data Mover) 2D tile load: global -> LDS, with LDS row padding.
// ---------------------------------------------------------------------------
#if defined(__gfx1250__) && __has_builtin(__builtin_amdgcn_tensor_load_to_lds)
#define HAVE_TDM_ 1
typedef unsigned int u32x4_ __attribute__((ext_vector_type(4)));
typedef int i32x8_ __attribute__((ext_vector_type(8)));
typedef int i32x4_ __attribute__((ext_vector_type(4)));
typedef const __attribute__((address_space(3))) void* lds_cptr_;

static __device__ __forceinline__ unsigned lds_off_(const void* p) {
    return (unsigned)(unsigned long long)(lds_cptr_)p;   // addrspacecast -> LDS byte offset
}

// rows x row_elems f32 tile; memory row stride = stride_elems floats.
// pad_int_code: pad every 2^(c+1) DWORDs ; pad_amt_code: insert (a+1) DWORDs.
static __device__ __forceinline__ void tdm_load_2d_(
    const void* gptr, unsigned ldsoff,
    unsigned rows, unsigned row_elems, unsigned stride_elems,
    unsigned pad_int_code, unsigned pad_amt_code)
{
    const unsigned long long ga = (unsigned long long)gptr;
    u32x4_ g0;
    g0[0] = 1u;                                         // count=1 (valid D#)
    g0[1] = ldsoff;                                     // lds_addr (bytes)
    g0[2] = (unsigned)(ga & 0xFFFFFFFFu);               // global_addr[31:0]
    g0[3] = (unsigned)((ga >> 32) & 0x01FFFFFFu) | (2u << 30); // [56:32] | type=2
    const unsigned td0 = 0x7FFFFFFFu, td1 = 0x7FFFFFFFu; // huge dims: no OOB clip
    i32x8_ g1;
    g1[0] = (int)((2u << 16) | (1u << 20) |             // data_size=4B, pad_enable
                  (pad_int_code << 22) | (pad_amt_code << 25));
    g1[1] = (int)((td0 & 0xFFFFu) << 16);               // tensor_dim0 lo16 (atomic addr=0)
    g1[2] = (int)((td0 >> 16) | ((td1 & 0xFFFFu) << 16));
    g1[3] = (int)((td1 >> 16) | ((row_elems & 0xFFFFu) << 16)); // tile_dim0
    g1[4] = (int)(rows & 0xFFFFu);                      // tile_dim1 (tile_dim2=0)
    g1[5] = (int)stride_elems;                          // tensor_dim0_stride[31:0]
    g1[6] = 0;
    g1[7] = 0;
    const i32x4_ z4 = {0, 0, 0, 0};                     // groups 2/3 unused (2D)
    const i32x8_ z8 = {0, 0, 0, 0, 0, 0, 0, 0};
    __builtin_amdgcn_tensor_load_to_lds(g0, g1, z4, z4, z8, 0);
}

static __device__ __forceinline__ void tdm_wait_() {
#if __has_builtin(__builtin_amdgcn_s_wait_tensorcnt)
    __builtin_amdgcn_s_wait_tensorcnt(0);
#else
    asm volatile("s_wait_tensorcnt 0x0" ::: "memory");
#endif
}
#else
#define HAVE_TDM_ 0
#endif

// ---------------------------------------------------------------------------
// K1: xn = emb_node*mask ; h_s = xn@Ws ; h_t = xn@Wt   (tiny)
// ---------------------------------------------------------------------------
__global__ __launch_bounds__(256)
void k1_node_proj(const float* __restrict__ emb_node,
                  const float* __restrict__ node_mask,
                  const float* __restrict__ Ws,
                  const float* __restrict__ Wt,
                  float* __restrict__ ws)
{
    __shared__ float xrow[F_];
    const int bi = blockIdx.x;           // b*N + i
    const int t  = threadIdx.x;
    const float m  = node_mask[bi];
    const float xv = emb_node[(size_t)bi * F_ + t] * m;
    xrow[t] = xv;
    ws[WS_XN + (size_t)bi * F_ + t] = xv;
    __syncthreads();
    float* hs = ws + WS_HS + (size_t)bi * HD_;
    float* ht = ws + WS_HT + (size_t)bi * HD_;
    for (int c = t; c < HD_; c += 256) {
        float as = 0.f, at = 0.f;
#pragma unroll 4
        for (int f = 0; f < F_; ++f) {
            const float x = xrow[f];
            as += x * Ws[f * HD_ + c];
            at += x * Wt[f * HD_ + c];
        }
        hs[c] = as;
        ht[c] = at;
    }
}

// ---------------------------------------------------------------------------
// K2: fused  h_e = (m_i*m_j)*(raw_xe@We) -> leakyrelu(h_s+h_t+h_e).attn
//     -> exp * adj * e_mask.  Mask factored out of the GEMM so panels are
//     raw strided copies (TDM-friendly). Deterministic shfl-tree reductions.
// ---------------------------------------------------------------------------
__global__ __launch_bounds__(256)
void k2_scores(const float* __restrict__ emb_edge,
               const float* __restrict__ adj,
               const float* __restrict__ node_mask,
               const float* __restrict__ We,
               const float* __restrict__ attn,
               float* __restrict__ ws)
{
    __shared__ float Ap[64][68];       // raw xe panel: 64 j x 64 k
    __shared__ float Bp[64][132];      // We panel: 64 k x 128 c
    __shared__ float scoreAcc[64][8];
    __shared__ float maskj[64];

    const int t    = threadIdx.x;
    const int wave = t >> 5, lane = t & 31;
    const int l15  = lane & 15, lhi = lane >> 4;
    const int bi   = blockIdx.x >> 2;          // b*N + i
    const int jt   = blockIdx.x & 3;
    const int b    = bi / N_;
    const int j0   = jt * 64;
    const float m_i = node_mask[bi];
    const int m_tile = wave & 3;
    const int whalf  = wave >> 2;

    if (t < 64) maskj[t] = node_mask[b * N_ + j0 + t];

    const float* hs      = ws + WS_HS + (size_t)bi * HD_;
    const float* ht_base = ws + WS_HT + (size_t)b * N_ * HD_;
    const float* Asrc    = emb_edge + (((size_t)bi) * N_ + j0) * F_;

    for (int c0 = 0; c0 < HD_; c0 += 128) {    // 4 column chunks of 128
        v8f acc[4];
#pragma unroll
        for (int nt = 0; nt < 4; ++nt)
            acc[nt] = (v8f){0.f,0.f,0.f,0.f,0.f,0.f,0.f,0.f};

        for (int p = 0; p < 4; ++p) {          // K panels of 64 (K=256)
            const int k0 = p * 64;
            __syncthreads();
#if HAVE_TDM_
            if (wave == 0) {
                tdm_load_2d_(Asrc + k0, lds_off_(&Ap[0][0]), 64, 64, F_, 5, 3);
                tdm_load_2d_(We + (size_t)k0 * HD_ + c0, lds_off_(&Bp[0][0]),
                             64, 128, HD_, 6, 3);
                tdm_wait_();
            }
#else
#pragma unroll
            for (int it = 0; it < 16; ++it) {
                const int idx = it * 256 + t;
                const int jj = idx >> 6, kk = idx & 63;
                Ap[jj][kk] = Asrc[(size_t)jj * F_ + k0 + kk];
            }
#pragma unroll
            for (int it = 0; it < 32; ++it) {
                const int idx = it * 256 + t;
                const int kk = idx >> 7, cc = idx & 127;
                Bp[kk][cc] = We[(size_t)(k0 + kk) * HD_ + c0 + cc];
            }
#endif
            __syncthreads();
            for (int ks = 0; ks < 16; ++ks) {  // 16 wmma k-steps of 4
                const int kb = ks * 4 + lhi * 2;
                v2f a;
                a.x = Ap[m_tile * 16 + l15][kb];
                a.y = Ap[m_tile * 16 + l15][kb + 1];
#pragma unroll
                for (int nt = 0; nt < 4; ++nt) {
                    const int cc = whalf * 64 + nt * 16 + l15;
                    v2f bb;
                    bb.x = Bp[kb][cc];
                    bb.y = Bp[kb + 1][cc];
                    acc[nt] = __builtin_amdgcn_wmma_f32_16x16x4_f32(
                        false, a, false, bb, (short)0, acc[nt], false, false);
                }
            }
        }
        // epilogue: this wave's 4 tiles live in ONE head
        const int h = (c0 >> 6) + whalf;
        float pr[8];
#pragma unroll
        for (int r = 0; r < 8; ++r) pr[r] = 0.f;
#pragma unroll
        for (int nt = 0; nt < 4; ++nt) {
            const int c   = c0 + whalf * 64 + nt * 16 + l15;
            const float av  = attn[c];
            const float hsv = hs[c];
#pragma unroll
            for (int r = 0; r < 8; ++r) {
                const int row = m_tile * 16 + r + lhi * 8;
                const int j   = j0 + row;
                const float em = m_i * maskj[row];     // mask applied post-GEMM
                float v = hsv + ht_base[(size_t)j * HD_ + c] + em * acc[nt][r];
                v = (v > 0.f) ? v : ALPHA_ * v;
                pr[r] += av * v;
            }
        }
#pragma unroll
        for (int r = 0; r < 8; ++r) {          // deterministic 16-lane tree
            float v = pr[r];
            v += __shfl_xor(v, 1, 32);
            v += __shfl_xor(v, 2, 32);
            v += __shfl_xor(v, 4, 32);
            v += __shfl_xor(v, 8, 32);
            if (l15 == 0)
                scoreAcc[m_tile * 16 + r + lhi * 8][h] = v;
        }
    }
    __syncthreads();
    for (int idx = t; idx < 512; idx += 256) {
        const int row = idx >> 3, hh = idx & 7;
        const int j = j0 + row;
        const float s = __expf(scoreAcc[row][hh])
                        * adj[((size_t)bi) * N_ + j] * m_i * maskj[row];
        ws[WS_SC + (((size_t)bi) * N_ + j) * H_ + hh] = s;
    }
}

// ---------------------------------------------------------------------------
// K3: softmax denoms, new_node = sum_jh s*h_t, @Wn+bn, residual+LN (node out)
//     + per-node eI = nn@Wedge[0:256]+bedge ; eJ = nn@Wedge[256:512]
// ---------------------------------------------------------------------------
__global__ __launch_bounds__(256)
void k3_node_out(const float* __restrict__ Wn,
                 const float* __restrict__ bn,
                 const float* __restrict__ Wedge,
                 const float* __restrict__ bedge,
                 const float* __restrict__ gamma_x,
                 const float* __restrict__ beta_x,
                 float* __restrict__ ws,
                 float* __restrict__ out)
{
    __shared__ float scn[N_ * H_];
    __shared__ float dpart[256];
    __shared__ float dinv[H_];
    __shared__ float ndpart[4][D_];
    __shared__ float nnd[D_];
    __shared__ float nnrow[F_];
    __shared__ float red[256], red2[256];
    __shared__ float stats[2];

    const int bi = blockIdx.x;
    const int b  = bi / N_;
    const int t  = threadIdx.x;
    const float* sc = ws + WS_SC + (size_t)bi * N_ * H_;

    { // denominators over j (fixed order)
        const int h = t & 7, jc = t >> 3;
        float p = 0.f;
        for (int q = 0; q < 8; ++q) p += sc[(jc * 8 + q) * H_ + h];
        dpart[t] = p;
    }
    __syncthreads();
    if (t < H_) {
        float s = 0.f;
        for (int q = 0; q < 32; ++q) s += dpart[q * 8 + t];
        dinv[t] = 1.f / (s + 1e-6f);
    }
    __syncthreads();
    for (int idx = t; idx < N_ * H_; idx += 256)
        scn[idx] = sc[idx] * dinv[idx & 7];
    __syncthreads();

    { // new_node_d[d] = sum_{j,h} scn * h_t
        const int d = t & 63, jc = t >> 6;
        const float* ht = ws + WS_HT + (size_t)b * N_ * HD_;
        float p = 0.f;
        for (int j = jc * 64; j < jc * 64 + 64; ++j) {
            const float* htj = ht + (size_t)j * HD_ + d;
            const float* sj  = scn + j * H_;
#pragma unroll
            for (int h = 0; h < H_; ++h) p += sj[h] * htj[h * D_];
        }
        ndpart[jc][d] = p;
    }
    __syncthreads();
    if (t < D_)
        nnd[t] = ndpart[0][t] + ndpart[1][t] + ndpart[2][t] + ndpart[3][t];
    __syncthreads();

    // new_node_in = nnd @ Wn + bn
    float v = bn[t];
#pragma unroll 4
    for (int d = 0; d < D_; ++d) v += nnd[d] * Wn[d * F_ + t];
    ws[WS_NN + (size_t)bi * F_ + t] = v;
    nnrow[t] = v;

    // out_node = LN(xn + new_node_in)
    const float y = ws[WS_XN + (size_t)bi * F_ + t] + v;
    red[t] = y; red2[t] = y * y;
    __syncthreads();
    for (int s = 128; s > 0; s >>= 1) {
        if (t < s) { red[t] += red[t + s]; red2[t] += red2[t + s]; }
        __syncthreads();
    }
    if (t == 0) {
        const float mean = red[0] * (1.f / F_);
        const float var  = red2[0] * (1.f / F_) - mean * mean;
        stats[0] = mean;
        stats[1] = rsqrtf(var + LN_EPS_);
    }
    __syncthreads();
    out[(size_t)bi * F_ + t] = (y - stats[0]) * stats[1] * gamma_x[t] + beta_x[t];

    // eI / eJ broadcast contributions for the edge GEMM (once per node)
    float ei = bedge[t], ej = 0.f;
#pragma unroll 4
    for (int k = 0; k < F_; ++k) {
        const float nk = nnrow[k];
        ei += nk * Wedge[k * F_ + t];
        ej += nk * Wedge[(F_ + k) * F_ + t];
    }
    ws[WS_EI + (size_t)bi * F_ + t] = ei;
    ws[WS_EJ + (size_t)bi * F_ + t] = ej;
}

// ---------------------------------------------------------------------------
// K4: edge GEMM, xe segment only (K=256):
//     y = em*(raw_xe@Wedge[512:768] + raw_xe) + eI[i] + eJ[j]
// Writes pre-LN y + deterministic per-row (sum,sumsq) stats.
// ---------------------------------------------------------------------------
__global__ __launch_bounds__(256)
void k4_edge_gemm(const float* __restrict__ emb_edge,
                  const float* __restrict__ node_mask,
                  const float* __restrict__ Wedge,
                  float* __restrict__ ws,
                  float* __restrict__ out)
{
    __shared__ float Ap[64][68];       // raw xe panel
    __shared__ float Bp[64][132];      // Wedge[512:768] panel
    __shared__ float rsum[64][4];
    __shared__ float rsq[64][4];
    __shared__ float maskj[64];

    const int t    = threadIdx.x;
    const int wave = t >> 5, lane = t & 31;
    const int l15  = lane & 15, lhi = lane >> 4;
    const int bi   = blockIdx.x >> 2;          // b*N + i
    const int jt   = blockIdx.x & 3;
    const int b    = bi / N_;
    const int j0   = jt * 64;
    const float m_i = node_mask[bi];
    const int m_tile = wave & 3;
    const int whalf  = wave >> 2;

    if (t < 64) maskj[t] = node_mask[b * N_ + j0 + t];

    const float* eI = ws + WS_EI + (size_t)bi * F_;
    const float* eJ = ws + WS_EJ + (size_t)b * N_ * F_;
    const float* Asrc = emb_edge + (((size_t)bi) * N_ + j0) * F_;
    float* edge_out = out + (size_t)B_ * N_ * F_;

    for (int half = 0; half < 2; ++half) {     // output cols in 2 chunks of 128
        const int c0 = half * 128;
        v8f acc[4];
#pragma unroll
        for (int nt = 0; nt < 4; ++nt)
            acc[nt] = (v8f){0.f,0.f,0.f,0.f,0.f,0.f,0.f,0.f};

        for (int p = 0; p < 4; ++p) {          // K=256 in panels of 64
            const int k0 = p * 64;
            __syncthreads();
#if HAVE_TDM_
            if (wave == 0) {
                tdm_load_2d_(Asrc + k0, lds_off_(&Ap[0][0]), 64, 64, F_, 5, 3);
                tdm_load_2d_(Wedge + (size_t)(2 * F_ + k0) * F_ + c0,
                             lds_off_(&Bp[0][0]), 64, 128, F_, 6, 3);
                tdm_wait_();
            }
#else
#pragma unroll
            for (int it = 0; it < 16; ++it) {
                const int idx = it * 256 + t;
                const int jj = idx >> 6, kk = idx & 63;
                Ap[jj][kk] = Asrc[(size_t)jj * F_ + k0 + kk];
            }
#pragma unroll
            for (int it = 0; it < 32; ++it) {
                const int idx = it * 256 + t;
                const int kk = idx >> 7, cc = idx & 127;
                Bp[kk][cc] = Wedge[(size_t)(2 * F_ + k0 + kk) * F_ + c0 + cc];
            }
#endif
            __syncthreads();
            for (int ks = 0; ks < 16; ++ks) {
                const int kb = ks * 4 + lhi * 2;
                v2f a;
                a.x = Ap[m_tile * 16 + l15][kb];
                a.y = Ap[m_tile * 16 + l15][kb + 1];
#pragma unroll
                for (int nt = 0; nt < 4; ++nt) {
                    const int cc = whalf * 64 + nt * 16 + l15;
                    v2f bb;
                    bb.x = Bp[kb][cc];
                    bb.y = Bp[kb + 1][cc];
                    acc[nt] = __builtin_amdgcn_wmma_f32_16x16x4_f32(
                        false, a, false, bb, (short)0, acc[nt], false, false);
                }
            }
        }
        // epilogue: y = em*(acc + xe_raw) + eI + eJ
        float psum[8], psq[8];
#pragma unroll
        for (int r = 0; r < 8; ++r) { psum[r] = 0.f; psq[r] = 0.f; }
#pragma unroll
        for (int nt = 0; nt < 4; ++nt) {
            const int c  = c0 + whalf * 64 + nt * 16 + l15;
            const float eic = eI[c];
#pragma unroll
            for (int r = 0; r < 8; ++r) {
                const int row = m_tile * 16 + r + lhi * 8;
                const int j   = j0 + row;
                const float em = m_i * maskj[row];
                const float xe = emb_edge[(((size_t)bi) * N_ + j) * F_ + c];
                const float y = em * (acc[nt][r] + xe) + eic
                                + eJ[(size_t)j * F_ + c];
                edge_out[(((size_t)bi) * N_ + j) * F_ + c] = y;
                psum[r] += y;
                psq[r]  += y * y;
            }
        }
#pragma unroll
        for (int r = 0; r < 8; ++r) {          // deterministic 16-lane tree
            float s = psum[r], q = psq[r];
            s += __shfl_xor(s, 1, 32);  q += __shfl_xor(q, 1, 32);
            s += __shfl_xor(s, 2, 32);  q += __shfl_xor(q, 2, 32);
            s += __shfl_xor(s, 4, 32);  q += __shfl_xor(q, 4, 32);
            s += __shfl_xor(s, 8, 32);  q += __shfl_xor(q, 8, 32);
            if (l15 == 0) {
                const int row = m_tile * 16 + r + lhi * 8;
                rsum[row][half * 2 + whalf] = s;   // each slot written once
                rsq[row][half * 2 + whalf]  = q;
            }
        }
    }
    __syncthreads();
    if (t < 64) {
        const size_t e = ((size_t)bi) * N_ + j0 + t;
        ws[WS_ESUM + e] = rsum[t][0] + rsum[t][1] + rsum[t][2] + rsum[t][3];
        ws[WS_ESQ  + e] = rsq[t][0]  + rsq[t][1]  + rsq[t][2]  + rsq[t][3];
    }
}

// ---------------------------------------------------------------------------
// K5: streaming edge LayerNorm using precomputed row stats (in-place on out)
// ---------------------------------------------------------------------------
__global__ __launch_bounds__(256)
void k5_edge_ln(const float* __restrict__ gamma_e,
                const float* __restrict__ beta_e,
                const float* __restrict__ ws,
                float* __restrict__ out)
{
    const size_t idx = (size_t)blockIdx.x * 256 + threadIdx.x;
    const int f = (int)(idx & (F_ - 1));
    const size_t e = idx >> 8;
    const float mean = ws[WS_ESUM + e] * (1.f / F_);
    const float var  = ws[WS_ESQ + e] * (1.f / F_) - mean * mean;
    float* edge = out + (size_t)B_ * N_ * F_;
    const float y = edge[idx];
    edge[idx] = (y - mean) * rsqrtf(var + LN_EPS_) * gamma_e[f] + beta_e[f];
}

// ---------------------------------------------------------------------------
extern "C" void kernel_launch(void* const* d_in, const int* in_sizes, int n_in,
                              void* d_out, int out_size, void* d_ws, size_t ws_size,
                              hipStream_t stream)
{
    const float* emb_node = (const float*)d_in[0];
    const float* emb_edge = (const float*)d_in[1];
    const float* adj      = (const float*)d_in[2];
    const float* mask     = (const float*)d_in[3];
    const float* Ws       = (const float*)d_in[4];
    const float* Wt       = (const float*)d_in[5];
    const float* We       = (const float*)d_in[6];
    const float* attn     = (const float*)d_in[7];
    const float* Wn       = (const float*)d_in[8];
    const float* bn       = (const float*)d_in[9];
    const float* Wedge    = (const float*)d_in[10];
    const float* bedge    = (const float*)d_in[11];
    const float* gx       = (const float*)d_in[12];
    const float* bx       = (const float*)d_in[13];
    const float* ge       = (const float*)d_in[14];
    const float* be       = (const float*)d_in[15];
    float* out = (float*)d_out;
    float* ws  = (float*)d_ws;

    k1_node_proj<<<B_ * N_,     256, 0, stream>>>(emb_node, mask, Ws, Wt, ws);
    k2_scores   <<<B_ * N_ * 4, 256, 0, stream>>>(emb_edge, adj, mask, We, attn, ws);
    k3_node_out <<<B_ * N_,     256, 0, stream>>>(Wn, bn, Wedge, bedge, gx, bx, ws, out);
    k4_edge_gemm<<<B_ * N_ * 4, 256, 0, stream>>>(emb_edge, mask, Wedge, ws, out);
    k5_edge_ln  <<<(B_ * N_ * N_ * F_) / 256, 256, 0, stream>>>(ge, be, ws, out);
}